// DeepGATGNN_47974784696360
// MI455X (gfx1250) — compile-verified
//
#include <hip/hip_runtime.h>

// ---------------------------------------------------------------------------
// DeepGATGNN forward for MI455X (gfx1250, wave32, WMMA f32).
// ---------------------------------------------------------------------------
constexpr int kN  = 10000;   // nodes
constexpr int kE  = 40000;   // edges
constexpr int kG  = 128;     // graphs
constexpr int kNF = 92;      // node feature dim
constexpr int kEF = 50;      // edge feature dim
constexpr int kEFp= 52;      // edge feature dim padded to %4
constexpr int kH  = 64;      // hidden
constexpr int kNH = 10;      // heads
constexpr int kGD = 108;     // global feature dim
constexpr int kL  = 5;       // layers
constexpr int kHH = kNH * kH; // 640

typedef __attribute__((ext_vector_type(2))) float v2f;
typedef __attribute__((ext_vector_type(8))) float v8f;

__device__ __forceinline__ float lrelu(float v) { return v > 0.f ? v : 0.2f * v; }

// ---------------------------------------------------------------------------
// Fast fp32 GEMM on the matrix pipe: C[M,NC] = A[M,K] @ B[K,NC]
// via V_WMMA_F32_16X16X4_F32.  Compile-time K and NC: fully unrolled chain of
// K/4 WMMAs, no bounds checks, all loads/stores use immediate offsets off a
// single base register.  Requires M%16==0, NC%16==0, K%4==0 (K even for
// 8-byte-aligned float2 A loads).  8 waves/block, one 16x16 tile per wave.
// EPI: 0 = raw, 1 = +bias then leaky_relu(0.2), 2 = +bias then relu
// ---------------------------------------------------------------------------
template <int K, int NC, int EPI>
__global__ void gemm_wmma_fast(const float* __restrict__ A, const float* __restrict__ B,
                               const float* __restrict__ bias, float* __restrict__ C,
                               int M)
{
  const int lane  = threadIdx.x & 31;
  const int wave  = threadIdx.x >> 5;
  const int tileM = blockIdx.y * 8 + wave;
  const int tileN = blockIdx.x;
  if (tileM * 16 >= M) return;                  // wave-uniform tail guard only

  const int m  = tileM * 16 + (lane & 15);
  const int n  = tileN * 16 + (lane & 15);
  const int kh = (lane >> 4) << 1;              // lanes 0-15: K{0,1}, 16-31: K{2,3}

  const float* __restrict__ Ap = A + (size_t)m * K + kh;   // 8B aligned (K even)
  const float* __restrict__ Bp = B + (size_t)kh * NC + n;

  v8f acc = {0.f, 0.f, 0.f, 0.f, 0.f, 0.f, 0.f, 0.f};
#pragma unroll
  for (int k0 = 0; k0 < K; k0 += 4) {
    v2f a = *(const v2f*)(Ap + k0);             // global_load_b64, imm offset
    v2f b;
    b.x = Bp[k0 * NC];                          // imm offsets (NC compile-time)
    b.y = Bp[(k0 + 1) * NC];
    acc = __builtin_amdgcn_wmma_f32_16x16x4_f32(false, a, false, b,
                                                (short)0, acc, false, false);
  }

  const int rbase = tileM * 16 + ((lane >> 4) << 3);
  const float bv = (EPI != 0) ? bias[n] : 0.f;
  float* __restrict__ Cp = C + (size_t)rbase * NC + n;
#pragma unroll
  for (int r = 0; r < 8; ++r) {
    float v = acc[r];
    if (EPI == 1)      v = lrelu(v + bv);
    else if (EPI == 2) v = fmaxf(v + bv, 0.f);
    Cp[r * NC] = v;                             // imm offsets
  }
}

// dst[Md,Kd] = src[Ms,Ks] zero-padded (row-major)
__global__ void pad2d(float* __restrict__ dst, const float* __restrict__ src,
                      int Md, int Kd, int Ms, int Ks)
{
  int i = blockIdx.x * blockDim.x + threadIdx.x;
  if (i >= Md * Kd) return;
  int r = i / Kd, c = i - r * Kd;
  dst[i] = (r < Ms && c < Ks) ? src[(size_t)r * Ks + c] : 0.f;
}

// ---------------------------------------------------------------------------
// Per-edge GAT attention. One wave per edge. Lane t owns columns 2t,2t+1 of
// each of the 10 heads (contiguous -> b64 gathers). Per-head attention dot via
// wave butterfly reduction, softmax over heads in registers, fused head-mean,
// fp32 atomic scatter into agg[row].
// ---------------------------------------------------------------------------
__global__ void edge_attn(const float* __restrict__ hW, const float* __restrict__ eaW,
                          const int* __restrict__ row, const int* __restrict__ col,
                          const float* __restrict__ att, const float* __restrict__ gamma,
                          const float* __restrict__ beta, float* __restrict__ agg, int nE)
{
  const int lane = threadIdx.x & 31;
  const int e    = blockIdx.x * (blockDim.x >> 5) + (threadIdx.x >> 5);
  if (e >= nE) return;

  const int r = row[e], c = col[e];
  const float* __restrict__ hr = hW  + (size_t)r * kHH + 2 * lane;
  const float* __restrict__ hc = hW  + (size_t)c * kHH + 2 * lane;
  const float* __restrict__ ew = eaW + (size_t)e * kHH + 2 * lane;
  const float* __restrict__ at = att + 2 * lane;

  float hj0[kNH], hj1[kNH], al[kNH];
#pragma unroll
  for (int h = 0; h < kNH; ++h) {
    const v2f ev = *(const v2f*)(ew + h * kH);
    const v2f rv = *(const v2f*)(hr + h * kH);
    const v2f cv = *(const v2f*)(hc + h * kH);
    const v2f aA = *(const v2f*)(at + h * 128);        // att[h, 0:64] slice
    const v2f aB = *(const v2f*)(at + h * 128 + kH);   // att[h, 64:128] slice
    const float i0 = lrelu(rv.x + ev.x), i1 = lrelu(rv.y + ev.y);
    const float j0 = lrelu(cv.x + ev.x), j1 = lrelu(cv.y + ev.y);
    hj0[h] = j0; hj1[h] = j1;
    al[h] = i0 * aA.x + i1 * aA.y + j0 * aB.x + j1 * aB.y;
  }
  // reduce each head's partial dot across the wave (all lanes end with the sum)
#pragma unroll
  for (int h = 0; h < kNH; ++h) {
    float p = al[h];
#pragma unroll
    for (int off = 16; off >= 1; off >>= 1) p += __shfl_xor(p, off, 32);
    al[h] = p;
  }
  // lrelu + DiffGroupNorm (eval) + softmax over heads
  const float inv_std = 0.99999500003749969f;   // 1/sqrt(1+1e-5)
  float mx = -1e30f;
#pragma unroll
  for (int h = 0; h < kNH; ++h) {
    float a = lrelu(al[h]);
    a = a * inv_std * gamma[h] + beta[h];
    al[h] = a;
    mx = fmaxf(mx, a);
  }
  float s = 0.f;
#pragma unroll
  for (int h = 0; h < kNH; ++h) { al[h] = __expf(al[h] - mx); s += al[h]; }
  const float inv = 1.0f / ((float)kNH * s);    // softmax denom * head-mean
  float m0 = 0.f, m1 = 0.f;
#pragma unroll
  for (int h = 0; h < kNH; ++h) { m0 += al[h] * hj0[h]; m1 += al[h] * hj1[h]; }
  float* dst = agg + (size_t)r * kH + 2 * lane;
  unsafeAtomicAdd(dst,     m0 * inv);
  unsafeAtomicAdd(dst + 1, m1 * inv);
}

// ---------------------------------------------------------------------------
// small elementwise helpers
// ---------------------------------------------------------------------------
__global__ void vzero(float* p, int n) {
  int i = blockIdx.x * blockDim.x + threadIdx.x;
  if (i < n) p[i] = 0.f;
}
__global__ void vcopy(float* dst, const float* src, int n) {
  int i = blockIdx.x * blockDim.x + threadIdx.x;
  if (i < n) dst[i] = src[i];
}
__global__ void vadd(float* dst, const float* src, int n) {
  int i = blockIdx.x * blockDim.x + threadIdx.x;
  if (i < n) dst[i] += src[i];
}
// h = (addOld ? h : 0) + agg + b[col]
__global__ void node_update(float* h, const float* agg, const float* b, int addOld, int n) {
  int i = blockIdx.x * blockDim.x + threadIdx.x;
  if (i >= n) return;
  float v = agg[i] + b[i & (kH - 1)];
  h[i] = (addOld ? h[i] : 0.f) + v;
}

// ---------------------------------------------------------------------------
// Global attention pooling
// ---------------------------------------------------------------------------
__device__ __forceinline__ unsigned fenc(float f) {
  unsigned b = __float_as_uint(f);
  return (b & 0x80000000u) ? ~b : (b | 0x80000000u);
}
__device__ __forceinline__ float fdec(unsigned u) {
  return __uint_as_float((u & 0x80000000u) ? (u & 0x7FFFFFFFu) : ~u);
}

// one wave per node: score = lrelu(t1[n]+t2[batch[n]]+b1) . W2 + b2 ; atomic segment max
__global__ void score_kernel(const float* __restrict__ t1, const float* __restrict__ t2,
                             const int* __restrict__ batch, const float* __restrict__ b1,
                             const float* __restrict__ W2, const float* __restrict__ b2,
                             float* __restrict__ score, unsigned* __restrict__ smax, int n)
{
  const int lane = threadIdx.x & 31;
  const int node = blockIdx.x * (blockDim.x >> 5) + (threadIdx.x >> 5);
  if (node >= n) return;
  const int g = batch[node];
  const v2f tv = *(const v2f*)(t1 + (size_t)node * kH + 2 * lane);
  const v2f gv = *(const v2f*)(t2 + (size_t)g * kH + 2 * lane);
  const v2f bb = *(const v2f*)(b1 + 2 * lane);
  const v2f wv = *(const v2f*)(W2 + 2 * lane);
  float p = lrelu(tv.x + gv.x + bb.x) * wv.x + lrelu(tv.y + gv.y + bb.y) * wv.y;
#pragma unroll
  for (int off = 16; off >= 1; off >>= 1) p += __shfl_xor(p, off, 32);
  if (lane == 0) {
    const float sc = p + b2[0];
    score[node] = sc;
    atomicMax(&smax[g], fenc(sc));
  }
}

__global__ void exp_kernel(const float* score, const int* batch, const unsigned* smax,
                           float* ex, float* denom, int n)
{
  int i = blockIdx.x * blockDim.x + threadIdx.x;
  if (i >= n) return;
  const int g = batch[i];
  const float v = __expf(score[i] - fdec(smax[g]));
  ex[i] = v;
  unsafeAtomicAdd(&denom[g], v);
}

__global__ void pool_kernel(const float* h, const int* batch, const float* ex,
                            const float* denom, float* pooled, int total)
{
  int i = blockIdx.x * blockDim.x + threadIdx.x;
  if (i >= total) return;
  const int node = i >> 6, d = i & (kH - 1);
  const int g = batch[node];
  const float w = ex[node] / denom[g];
  unsafeAtomicAdd(&pooled[(size_t)g * kH + d], h[i] * w);
}

// one wave per graph: out[g] = q[g] . W2 + b2
__global__ void out_kernel(const float* __restrict__ q, const float* __restrict__ W2,
                           const float* __restrict__ b2, float* __restrict__ out, int ng)
{
  const int lane = threadIdx.x & 31;
  const int g = blockIdx.x * (blockDim.x >> 5) + (threadIdx.x >> 5);
  if (g >= ng) return;
  const v2f qv = *(const v2f*)(q + (size_t)g * kH + 2 * lane);
  const v2f wv = *(const v2f*)(W2 + 2 * lane);
  float p = qv.x * wv.x + qv.y * wv.y;
#pragma unroll
  for (int off = 16; off >= 1; off >>= 1) p += __shfl_xor(p, off, 32);
  if (lane == 0) out[g] = p + b2[0];
}

// ---------------------------------------------------------------------------
extern "C" void kernel_launch(void* const* d_in, const int* in_sizes, int n_in,
                              void* d_out, int out_size, void* d_ws, size_t ws_size,
                              hipStream_t stream)
{
  (void)in_sizes; (void)n_in; (void)out_size; (void)ws_size;

  const float* x        = (const float*)d_in[0];
  const int*   eidx     = (const int*)  d_in[1];
  const float* eattr    = (const float*)d_in[2];
  const int*   batch    = (const int*)  d_in[3];
  const float* gfeat    = (const float*)d_in[4];
  const float* node_W   = (const float*)d_in[5];
  const float* node_b   = (const float*)d_in[6];
  const float* edge_W   = (const float*)d_in[7];
  const float* edge_b   = (const float*)d_in[8];
  const float* conv_W   = (const float*)d_in[9];
  const float* conv_att = (const float*)d_in[10];
  const float* conv_b   = (const float*)d_in[11];
  const float* conv_g   = (const float*)d_in[12];
  const float* conv_be  = (const float*)d_in[13];
  const float* ga_W1    = (const float*)d_in[14];
  const float* ga_b1    = (const float*)d_in[15];
  const float* ga_W2    = (const float*)d_in[16];
  const float* ga_b2    = (const float*)d_in[17];
  const float* out_W1   = (const float*)d_in[18];
  const float* out_b1   = (const float*)d_in[19];
  const float* out_W2   = (const float*)d_in[20];
  const float* out_b2   = (const float*)d_in[21];

  const int* row = eidx;       // edge_index[0]
  const int* col = eidx + kE;  // edge_index[1]

  // ---- workspace carving (peak ~155 MB; tail buffers alias hW/eaW) ----
  char* wsp = (char*)d_ws;
  size_t off = 0;
  auto carve = [&](size_t bytes) -> void* {
    void* p = wsp + off;
    off = (off + bytes + 255) & ~(size_t)255;
    return p;
  };
  float* h     = (float*)carve((size_t)kN * kH * 4);
  float* init  = (float*)carve((size_t)kN * kH * 4);
  float* ea    = (float*)carve((size_t)kE * kH * 4);
  float* agg   = (float*)carve((size_t)kN * kH * 4);
  float* eapad = (float*)carve((size_t)kE * kEFp * 4);   // edge_attr padded K=52
  float* eWpad = (float*)carve((size_t)kEFp * kH * 4);   // edge_W padded to 52 rows
  float* hW    = (float*)carve((size_t)kN * kHH * 4);
  float* eaW   = (float*)carve((size_t)kE * kHH * 4);

  // tail buffers, live only after the layer loop -> alias hW/eaW storage
  float*    t1     = hW;
  char*     tail   = (char*)eaW;
  auto tcarve = [&](size_t bytes) -> void* {
    void* p = tail; tail += (bytes + 255) & ~(size_t)255; return p;
  };
  float*    t2     = (float*)tcarve((size_t)kG * kH * 4);
  float*    score  = (float*)tcarve((size_t)kN * 4);
  float*    ex     = (float*)tcarve((size_t)kN * 4);
  unsigned* smax   = (unsigned*)tcarve((size_t)kG * 4);
  float*    denom  = (float*)tcarve((size_t)kG * 4);
  float*    pooled = (float*)tcarve((size_t)kG * kH * 4);
  float*    q      = (float*)tcarve((size_t)kG * kH * 4);

  auto grid_for = [](int M, int Nc) {
    return dim3((unsigned)(Nc / 16), (unsigned)((M / 16 + 7) / 8));
  };

  // ---- embeddings ----
  // h = lrelu(x @ node_W + b)      [10000,92]@[92,64]
  gemm_wmma_fast<kNF, kH, 1><<<grid_for(kN, kH), 256, 0, stream>>>(x, node_W, node_b, h, kN);
  // pad edge_attr 50->52 cols and edge_W 50->52 rows, then
  // ea = lrelu(edge_attr @ edge_W + b)   [40000,52]@[52,64]
  pad2d<<<(kE * kEFp + 255) / 256, 256, 0, stream>>>(eapad, eattr, kE, kEFp, kE, kEF);
  pad2d<<<(kEFp * kH + 255) / 256, 256, 0, stream>>>(eWpad, edge_W, kEFp, kH, kEF, kH);
  gemm_wmma_fast<kEFp, kH, 1><<<grid_for(kE, kH), 256, 0, stream>>>(eapad, eWpad, edge_b, ea, kE);
  vcopy<<<(kN * kH + 255) / 256, 256, 0, stream>>>(init, h, kN * kH);

  // ---- GAT layers ----
  for (int i = 0; i < kL; ++i) {
    const float* Wi   = conv_W + (size_t)i * 2 * kH * kHH;   // [128,640]
    const float* Wtop = Wi;                                  // rows 0..63 (node part)
    const float* Wbot = Wi + (size_t)kH * kHH;               // rows 64..127 (edge part)
    gemm_wmma_fast<kH, kHH, 0><<<grid_for(kN, kHH), 256, 0, stream>>>(h,  Wtop, nullptr, hW,  kN);
    gemm_wmma_fast<kH, kHH, 0><<<grid_for(kE, kHH), 256, 0, stream>>>(ea, Wbot, nullptr, eaW, kE);
    vzero<<<(kN * kH + 255) / 256, 256, 0, stream>>>(agg, kN * kH);
    edge_attn<<<(kE + 7) / 8, 256, 0, stream>>>(
        hW, eaW, row, col,
        conv_att + (size_t)i * kNH * 2 * kH,
        conv_g + i * kNH, conv_be + i * kNH, agg, kE);
    node_update<<<(kN * kH + 255) / 256, 256, 0, stream>>>(
        h, agg, conv_b + i * kH, (i > 0) ? 1 : 0, kN * kH);
  }
  vadd<<<(kN * kH + 255) / 256, 256, 0, stream>>>(h, init, kN * kH);  // h += initial

  // ---- global attention pooling ----
  // t1 = h @ ga_W1[:64]            [10000,64]@[64,64]
  gemm_wmma_fast<kH, kH, 0><<<grid_for(kN, kH), 256, 0, stream>>>(h, ga_W1, nullptr, t1, kN);
  // t2 = global_features @ ga_W1[64:172]   [128,108]@[108,64]
  gemm_wmma_fast<kGD, kH, 0><<<grid_for(kG, kH), 256, 0, stream>>>(
      gfeat, ga_W1 + (size_t)kH * kH, nullptr, t2, kG);
  vzero<<<1, 256, 0, stream>>>((float*)smax, kG);     // fenc(x) > 0 for all finite x
  vzero<<<1, 256, 0, stream>>>(denom, kG);
  vzero<<<(kG * kH + 255) / 256, 256, 0, stream>>>(pooled, kG * kH);
  score_kernel<<<(kN + 7) / 8, 256, 0, stream>>>(t1, t2, batch, ga_b1, ga_W2, ga_b2,
                                                 score, smax, kN);
  exp_kernel<<<(kN + 255) / 256, 256, 0, stream>>>(score, batch, smax, ex, denom, kN);
  pool_kernel<<<(kN * kH + 255) / 256, 256, 0, stream>>>(h, batch, ex, denom, pooled,
                                                         kN * kH);

  // ---- output MLP ----
  // q = relu(pooled @ out_W1 + b1)   [128,64]@[64,64]
  gemm_wmma_fast<kH, kH, 2><<<grid_for(kG, kH), 256, 0, stream>>>(pooled, out_W1, out_b1, q, kG);
  out_kernel<<<(kG + 7) / 8, 256, 0, stream>>>(q, out_W2, out_b2, (float*)d_out, kG);
}